// decoder_15015205666921
// MI455X (gfx1250) — compile-verified
//
#include <hip/hip_runtime.h>
#include <cstdint>

#define HIDDEN 1024
#define VOCAB  128000
#define ROWS_PER_BLOCK 128
#define TILE_K 64
#define LDSW_STRIDE 68   // 64 + 4 floats pad -> conflict-free WMMA A-operand fetch

typedef float v2f __attribute__((ext_vector_type(2)));
typedef float v4f __attribute__((ext_vector_type(4)));
typedef float v8f __attribute__((ext_vector_type(8)));

// gfx1250 async global->LDS (ASYNCcnt-tracked) if the toolchain exposes it.
#if defined(__AMDGCN__) && __has_builtin(__builtin_amdgcn_global_load_async_to_lds_b128) && __has_builtin(__builtin_amdgcn_s_wait_asynccnt)
#define HAVE_ASYNC_LDS 1
// Exact parameter types per hipcc diagnostic: v4i in AS(1) / AS(3), non-const.
typedef int v4i_gcc __attribute__((vector_size(4 * sizeof(int))));
typedef __attribute__((address_space(1))) v4i_gcc* as1_v4i_ptr;
typedef __attribute__((address_space(3))) v4i_gcc* as3_v4i_ptr;
#else
#define HAVE_ASYNC_LDS 0
#endif

// ---------------------------------------------------------------------------
// y[r] = sum_k W[r,k]*vec[k] + bias[r], 128 rows per block (16 per wave),
// K=1024.  W tile staged to LDS (async b128 copies), WMMA f32 16x16x4 with
// the vector broadcast across all 16 N columns (every D column == y).
// Optionally emits per-block (max, sum-exp) partials for log-softmax.
// ---------------------------------------------------------------------------
__global__ __launch_bounds__(256) void matvec_wmma_kernel(
    const float* __restrict__ W,
    const float* __restrict__ vec,
    const float* __restrict__ bias,
    float* __restrict__ out,
    float* __restrict__ partials,
    int do_partials)
{
  __shared__ float sW[ROWS_PER_BLOCK * LDSW_STRIDE]; // 34.0 KB weight tile
  __shared__ float sv[HIDDEN];                       // 4 KB vector
  __shared__ float slog[ROWS_PER_BLOCK];
  __shared__ float sred[ROWS_PER_BLOCK];

  const int tid  = threadIdx.x;
  const int wave = tid >> 5;
  const int lane = tid & 31;
  const long long rbase = (long long)blockIdx.x * ROWS_PER_BLOCK;

  for (int i = tid; i < HIDDEN; i += 256) sv[i] = vec[i];

  v8f acc = {};

  // A-matrix 16x4 f32 layout (ISA 7.12.2): lanes 0-15 -> M=lane, K=0,1;
  // lanes 16-31 -> M=lane-16, K=2,3.  B 4x16: lanes hold K=0,1 / K=2,3.
  const int mrow = wave * 16 + (lane & 15);
  const int ksub = (lane >> 4) << 1;  // 0 or 2

  for (int kb = 0; kb < HIDDEN; kb += TILE_K) {
    __syncthreads();  // previous tile fully consumed
    // Stage 128 rows x 64 K: 2048 float4, 8 per thread, 256 B contiguous/row.
#pragma unroll
    for (int i = 0; i < 8; ++i) {
      int f4  = tid + i * 256;
      int row = f4 >> 4;          // 16 float4 per row
      int col = (f4 & 15) << 2;   // float offset within row
      const float* gp = W + (rbase + row) * HIDDEN + kb + col;
      float*       lp = &sW[row * LDSW_STRIDE + col];
#if HAVE_ASYNC_LDS
      __builtin_amdgcn_global_load_async_to_lds_b128(
          (as1_v4i_ptr)(uintptr_t)gp,
          (as3_v4i_ptr)(uint32_t)(uintptr_t)lp,
          0, 0);
#else
      *(v4f*)lp = *(const v4f*)gp;
#endif
    }
#if HAVE_ASYNC_LDS
    __builtin_amdgcn_s_wait_asynccnt(0);
#endif
    __syncthreads();

#pragma unroll
    for (int kk = 0; kk < TILE_K; kk += 4) {
      v2f a, b;
      const int sa = mrow * LDSW_STRIDE + kk + ksub; // banks 4*lane.. : no conflicts
      a.x = sW[sa];
      a.y = sW[sa + 1];
      const int sb = kb + kk + ksub;                 // broadcast read
      b.x = sv[sb];
      b.y = sv[sb + 1];
      acc = __builtin_amdgcn_wmma_f32_16x16x4_f32(false, a, false, b,
                                                  (short)0, acc, false, false);
    }
  }

  // D layout: lanes 0-15 hold M=0..7 in acc[0..7]; lanes 16-31 hold M=8..15.
  const int half = lane >> 4;
  if ((lane & 15) == 0) {
#pragma unroll
    for (int j = 0; j < 8; ++j) {
      int rloc = wave * 16 + half * 8 + j;
      long long r = rbase + rloc;
      float val = acc[j] + bias[r];
      out[r]     = val;
      slog[rloc] = val;
    }
  }

  if (do_partials) {
    __syncthreads();
    if (tid < ROWS_PER_BLOCK) sred[tid] = slog[tid];
    for (int s = ROWS_PER_BLOCK / 2; s > 0; s >>= 1) {
      __syncthreads();
      if (tid < s) sred[tid] = fmaxf(sred[tid], sred[tid + s]);
    }
    __syncthreads();
    float M = sred[0];
    __syncthreads();
    if (tid < ROWS_PER_BLOCK) sred[tid] = expf(slog[tid] - M);
    for (int s = ROWS_PER_BLOCK / 2; s > 0; s >>= 1) {
      __syncthreads();
      if (tid < s) sred[tid] += sred[tid + s];
    }
    __syncthreads();
    if (tid == 0) {
      partials[2 * blockIdx.x]     = M;
      partials[2 * blockIdx.x + 1] = sred[0];
    }
  }
}

__global__ __launch_bounds__(256) void embed_relu_kernel(
    const int* __restrict__ tok, const float* __restrict__ emb,
    float* __restrict__ x)
{
  int k = blockIdx.x * 256 + threadIdx.x;
  long long t = (long long)tok[0];
  x[k] = fmaxf(emb[t * HIDDEN + k], 0.0f);
}

__global__ __launch_bounds__(1024) void gru_gates_kernel(
    const float* __restrict__ gi, const float* __restrict__ gh,
    const float* __restrict__ h,
    float* __restrict__ hnew_ws, float* __restrict__ hnew_out)
{
  int k = threadIdx.x;
  float r = 1.0f / (1.0f + expf(-(gi[k] + gh[k])));
  float z = 1.0f / (1.0f + expf(-(gi[HIDDEN + k] + gh[HIDDEN + k])));
  float n = tanhf(gi[2 * HIDDEN + k] + r * gh[2 * HIDDEN + k]);
  float hn = (1.0f - z) * n + z * h[k];
  hnew_ws[k]  = hn;
  hnew_out[k] = hn;
}

__global__ __launch_bounds__(1024) void logz_kernel(
    const float* __restrict__ partials, float* __restrict__ logz, int np)
{
  __shared__ float sA[1024];
  int tid = threadIdx.x;
  float Mb = -__builtin_inff(), Sb = 0.0f;
  if (tid < np) { Mb = partials[2 * tid]; Sb = partials[2 * tid + 1]; }
  sA[tid] = Mb;
  for (int s = 512; s > 0; s >>= 1) {
    __syncthreads();
    if (tid < s) sA[tid] = fmaxf(sA[tid], sA[tid + s]);
  }
  __syncthreads();
  float M = sA[0];
  __syncthreads();
  sA[tid] = (tid < np) ? Sb * expf(Mb - M) : 0.0f;
  for (int s = 512; s > 0; s >>= 1) {
    __syncthreads();
    if (tid < s) sA[tid] += sA[tid + s];
  }
  __syncthreads();
  if (tid == 0) logz[0] = M + logf(sA[0]);
}

__global__ __launch_bounds__(256) void sub_logz_kernel(
    float* __restrict__ out, const float* __restrict__ logz)
{
  int i = blockIdx.x * 256 + threadIdx.x;
  out[i] -= logz[0];
}

// ---------------------------------------------------------------------------
extern "C" void kernel_launch(void* const* d_in, const int* in_sizes, int n_in,
                              void* d_out, int out_size, void* d_ws, size_t ws_size,
                              hipStream_t stream) {
  (void)in_sizes; (void)n_in; (void)out_size; (void)ws_size;
  const int*   tok   = (const int*)  d_in[0];
  const float* hid   = (const float*)d_in[1];
  const float* emb   = (const float*)d_in[2];
  const float* w_ih  = (const float*)d_in[3];
  const float* w_hh  = (const float*)d_in[4];
  const float* b_ih  = (const float*)d_in[5];
  const float* b_hh  = (const float*)d_in[6];
  const float* w_out = (const float*)d_in[7];
  const float* b_out = (const float*)d_in[8];
  float* out = (float*)d_out;         // [0,VOCAB): logprobs, [VOCAB,VOCAB+H): h_new

  float* ws    = (float*)d_ws;
  float* x     = ws;            // 1024
  float* gi    = ws + 1024;     // 3072
  float* gh    = ws + 4096;     // 3072
  float* hnew  = ws + 7168;     // 1024
  float* parts = ws + 8192;     // 2 * 1000
  float* logz  = ws + 10192;    // 1

  embed_relu_kernel<<<HIDDEN / 256, 256, 0, stream>>>(tok, emb, x);
  matvec_wmma_kernel<<<3 * HIDDEN / ROWS_PER_BLOCK, 256, 0, stream>>>(
      w_ih, x, b_ih, gi, nullptr, 0);
  matvec_wmma_kernel<<<3 * HIDDEN / ROWS_PER_BLOCK, 256, 0, stream>>>(
      w_hh, hid, b_hh, gh, nullptr, 0);
  gru_gates_kernel<<<1, HIDDEN, 0, stream>>>(gi, gh, hid, hnew, out + VOCAB);
  matvec_wmma_kernel<<<VOCAB / ROWS_PER_BLOCK, 256, 0, stream>>>(
      w_out, hnew, b_out, out, parts, 1);
  logz_kernel<<<1, 1024, 0, stream>>>(parts, logz, VOCAB / ROWS_PER_BLOCK);
  sub_logz_kernel<<<VOCAB / 256, 256, 0, stream>>>(out, logz);
}